// SelectiveSSM_37374805409921
// MI455X (gfx1250) — compile-verified
//
#include <hip/hip_runtime.h>

typedef __attribute__((ext_vector_type(2))) float v2f;
typedef __attribute__((ext_vector_type(8))) float v8f;

#define D_MODEL 2048
#define D_STATE 16
#define DT_RANK 64
#define BATCH   4
#define SEQ     4096

// ---------------------------------------------------------------------------
// Kernel 1: x_proj = x @ W_xproj^T   (M=16384, K=2048, N=96) via f32 WMMA.
// One wave per 16-row M-tile; it computes all six 16-wide N-tiles so each x
// element is loaded exactly once (HBM floor ~5.5us for the 128MB x stream).
// WMMA f32 16x16x4 layouts (CDNA5 ISA 7.12.2):
//   A (16x4):  lanes 0-15 -> M=lane, K={0,1}; lanes 16-31 -> M=lane-16, K={2,3}
//   B (4x16):  v0: K=0 (lanes 0-15) / K=2 (lanes 16-31); v1: K=1 / K=3
//   C (16x16): VGPR v -> M=v (lanes 0-15), M=v+8 (lanes 16-31), N=lane&15
// Both A and B fragments are contiguous float2 loads per lane.
// ---------------------------------------------------------------------------
__global__ __launch_bounds__(256) void xproj_kernel(
    const float* __restrict__ x,        // (16384, 2048)
    const float* __restrict__ W,        // W_xproj (96, 2048)
    float* __restrict__ dtp_ws,         // (16384, 64)
    float* __restrict__ B_ws,           // (16384, 16)
    float* __restrict__ C_ws)           // (16384, 16)
{
    const int lane = threadIdx.x & 31;
    const int wave = threadIdx.x >> 5;
    const int mtile = blockIdx.x * 8 + wave;     // 1024 M-tiles total
    const int m0   = mtile * 16;
    const int half = lane >> 4;                  // 0 or 1
    const int l15  = lane & 15;
    const int koff = half * 2;

    const float* xrow = x + (size_t)(m0 + l15) * D_MODEL + koff;

    v8f acc[6];
#pragma unroll
    for (int t = 0; t < 6; ++t) acc[t] = (v8f){};

    for (int k = 0; k < D_MODEL; k += 4) {
        v2f a = *(const v2f*)(xrow + k);
#pragma unroll
        for (int t = 0; t < 6; ++t) {
            // B[k][n] = W_xproj[n][k]  (W row n is contiguous over k)
            v2f b = *(const v2f*)(W + (size_t)(t * 16 + l15) * D_MODEL + k + koff);
            acc[t] = __builtin_amdgcn_wmma_f32_16x16x4_f32(
                false, a, false, b, (short)0, acc[t], false, false);
        }
    }

    // Scatter C-layout accumulators into dtp / B / C workspace slices.
#pragma unroll
    for (int t = 0; t < 6; ++t) {
#pragma unroll
        for (int v = 0; v < 8; ++v) {
            const int m = m0 + v + half * 8;
            const float val = acc[t][v];
            if (t < 4)       dtp_ws[(size_t)m * DT_RANK + t * 16 + l15] = val;
            else if (t == 4) B_ws[(size_t)m * D_STATE + l15] = val;
            else             C_ws[(size_t)m * D_STATE + l15] = val;
        }
    }
}

// ---------------------------------------------------------------------------
// Kernel 2: fused dt-GEMM (WMMA) + softplus + selective scan + D*x epilogue,
// now software-pipelined: with only ~1.3 waves/SIMD the wave cannot rely on
// co-resident waves to hide memory latency, so each 16-step chunk prefetches
// the NEXT chunk's dtp A-fragments (32 VGPRs) and x values (16 VGPRs) into a
// second register buffer before running the current chunk's WMMA->softplus->
// scan body (~800 cycles of work, fully covering L2/HBM load latency).
// Wave layout: one wave owns (batch b, 16 channels); lane L / L^16 split the
// 16 states of channel (L&15). dt chunk is staged through a 1KB per-wave LDS
// slice to undo the WMMA C-layout (same-wave DS ops are in-order).
// ---------------------------------------------------------------------------
__global__ __launch_bounds__(128) void scan_kernel(
    const float* __restrict__ x,        // (4, 4096, 2048)
    const float* __restrict__ A_log,    // (2048, 16)
    const float* __restrict__ Dvec,     // (2048,)
    const float* __restrict__ W_dt,     // (2048, 64)
    const float* __restrict__ b_dt,     // (2048,)
    const float* __restrict__ dtp_ws,   // (16384, 64)
    const float* __restrict__ B_ws,     // (16384, 16)
    const float* __restrict__ C_ws,     // (16384, 16)
    float* __restrict__ out)            // (4, 4096, 2048)
{
    __shared__ float lds_dt[4][16 * 16];           // per-wave 1KB slice

    const int lane  = threadIdx.x & 31;
    const int wave  = threadIdx.x >> 5;
    const int flatW = blockIdx.x * 4 + wave;       // 512 waves
    const int bIdx  = flatW >> 7;                  // 128 d-tiles per batch
    const int d0    = (flatW & 127) * 16;
    const int half  = lane >> 4;
    const int l15   = lane & 15;
    const int koff  = half * 2;
    const int d     = d0 + l15;                    // channel for this lane pair
    const int nbase = half * 8;                    // state sub-range

    // Loop-invariant W_dt B-fragments: Bmat[k][n] = W_dt[d0+n][k]
    v2f wfrag[16];
#pragma unroll
    for (int kk = 0; kk < 16; ++kk)
        wfrag[kk] = *(const v2f*)(W_dt + (size_t)d * DT_RANK + kk * 4 + koff);

    float Areg[8];
#pragma unroll
    for (int j = 0; j < 8; ++j)
        Areg[j] = -__expf(A_log[(size_t)d * D_STATE + nbase + j]);

    const float bias = b_dt[d];
    const float Dd   = Dvec[d];

    float h[8];
#pragma unroll
    for (int j = 0; j < 8; ++j) h[j] = 0.0f;

    float* myLds = &lds_dt[wave][0];
    const size_t xbase  = (size_t)bIdx * SEQ * D_MODEL + d;
    const size_t bcbase = (size_t)bIdx * SEQ * D_STATE;
    const size_t dtpb   = (size_t)bIdx * SEQ * DT_RANK;

    // Double-buffered chunk inputs.
    v2f   afA[16], afB[16];
    float xvA[16], xvB[16];

    auto prefetch = [&](int ln, v2f (&an)[16], float (&xn)[16]) {
        const float* arown = dtp_ws + dtpb + (size_t)(ln + l15) * DT_RANK + koff;
#pragma unroll
        for (int kk = 0; kk < 16; ++kk)
            an[kk] = *(const v2f*)(arown + kk * 4);
#pragma unroll
        for (int t = 0; t < 16; ++t)
            xn[t] = x[xbase + (size_t)(ln + t) * D_MODEL];
    };

    auto do_chunk = [&](int l0, v2f (&ac)[16], float (&xc)[16]) {
        // dt tile via WMMA: (16 timesteps) x (16 channels), K=64
        v8f acc = (v8f){};
#pragma unroll
        for (int kk = 0; kk < 16; ++kk)
            acc = __builtin_amdgcn_wmma_f32_16x16x4_f32(
                false, ac[kk], false, wfrag[kk], (short)0, acc, false, false);

        // softplus in registers, scatter (timestep, channel) into LDS
#pragma unroll
        for (int v = 0; v < 8; ++v) {
            const float z  = acc[v] + bias;
            const float dt = (z > 20.0f) ? z : __logf(1.0f + __expf(z));
            myLds[(v + half * 8) * 16 + l15] = dt;
        }

        // serial scan over the chunk
#pragma unroll
        for (int t = 0; t < 16; ++t) {
            const int l = l0 + t;
            const float dt  = myLds[t * 16 + l15];      // broadcast to pair
            const float dtx = dt * xc[t];
            const float* Bp = B_ws + bcbase + (size_t)l * D_STATE + nbase;
            const float* Cp = C_ws + bcbase + (size_t)l * D_STATE + nbase;
            float y = 0.0f;
#pragma unroll
            for (int j = 0; j < 8; ++j) {
                const float Ab = __expf(dt * Areg[j]);  // A < 0 -> (0,1]
                h[j] = fmaf(Ab, h[j], dtx * Bp[j]);
                y    = fmaf(h[j], Cp[j], y);
            }
            y += __shfl_xor(y, 16, 32);                 // join state halves
            if (half == 0)
                out[xbase + (size_t)l * D_MODEL] = fmaf(Dd, xc[t], y);
        }
    };

    prefetch(0, afA, xvA);
    for (int l0 = 0; l0 < SEQ; l0 += 32) {
        prefetch(l0 + 16, afB, xvB);    // covers chunk A's body
        do_chunk(l0, afA, xvA);
        int ln = l0 + 32;
        if (ln >= SEQ) ln = SEQ - 16;   // clamped dummy prefetch on last iter
        prefetch(ln, afA, xvA);         // covers chunk B's body
        do_chunk(l0 + 16, afB, xvB);
    }
}

extern "C" void kernel_launch(void* const* d_in, const int* in_sizes, int n_in,
                              void* d_out, int out_size, void* d_ws, size_t ws_size,
                              hipStream_t stream) {
    (void)in_sizes; (void)n_in; (void)out_size; (void)ws_size;
    const float* x       = (const float*)d_in[0];
    const float* A_log   = (const float*)d_in[1];
    const float* Dv      = (const float*)d_in[2];
    const float* W_xproj = (const float*)d_in[3];
    const float* W_dt    = (const float*)d_in[4];
    const float* b_dt    = (const float*)d_in[5];
    float* out = (float*)d_out;

    float* dtp_ws = (float*)d_ws;                                  // 4 MB
    float* B_ws   = dtp_ws + (size_t)BATCH * SEQ * DT_RANK;        // 1 MB
    float* C_ws   = B_ws   + (size_t)BATCH * SEQ * D_STATE;        // 1 MB

    // 1024 M-tiles / 8 waves per block
    xproj_kernel<<<128, 256, 0, stream>>>(x, W_xproj, dtp_ws, B_ws, C_ws);
    // 512 waves / 4 waves per block
    scan_kernel<<<128, 128, 0, stream>>>(x, A_log, Dv, W_dt, b_dt,
                                         dtp_ws, B_ws, C_ws, out);
}